// GraphEncoder_13718125543797
// MI455X (gfx1250) — compile-verified
//
#include <hip/hip_runtime.h>
#include <hip/hip_bf16.h>

typedef float v2f __attribute__((ext_vector_type(2)));
typedef float v8f __attribute__((ext_vector_type(8)));

#define HID 128
#define WSTRIDE 130  // padded LDS row stride (floats): kills bank conflicts,
                     // keeps 8B alignment for ds_load_b64 (130*4=520=8*65)

// ---------------------------------------------------------------------------
// agg[i] = h[i]  (vectorized copy; "h +" part of GIN eps=0 aggregation)
// ---------------------------------------------------------------------------
__global__ void copy_f4_kernel(const float* __restrict__ in,
                               float* __restrict__ out, long long n4) {
  long long t = (long long)blockIdx.x * blockDim.x + threadIdx.x;
  if (t >= n4) return;
  reinterpret_cast<float4*>(out)[t] = reinterpret_cast<const float4*>(in)[t];
}

__global__ void zero_kernel(float* __restrict__ p, long long n) {
  long long t = (long long)blockIdx.x * blockDim.x + threadIdx.x;
  if (t < n) p[t] = 0.0f;
}

// ---------------------------------------------------------------------------
// Edge scatter:  agg[dst[e]][c] += h[src[e]][c]
// 32 threads per edge, 4 channels each (coalesced float4 gather, L2-resident:
// the whole 51.2MB feature matrix fits in the 192MB L2)
// ---------------------------------------------------------------------------
__global__ void edge_scatter_kernel(const int* __restrict__ src,
                                    const int* __restrict__ dst,
                                    const float* __restrict__ h,
                                    float* __restrict__ agg, int nEdges) {
  long long t = (long long)blockIdx.x * blockDim.x + threadIdx.x;
  long long total = (long long)nEdges * 32;
  if (t >= total) return;
  int e  = (int)(t >> 5);
  int c4 = (int)(t & 31) << 2;
  int s = src[e];
  int d = dst[e];
  float4 v = *reinterpret_cast<const float4*>(&h[(size_t)s * HID + c4]);
  float* p = &agg[(size_t)d * HID + c4];
  atomicAdd(p + 0, v.x);
  atomicAdd(p + 1, v.y);
  atomicAdd(p + 2, v.z);
  atomicAdd(p + 3, v.w);
}

// ---------------------------------------------------------------------------
// Graph pooling scatter: g[batch[v]][c] += h[v][c]
// ---------------------------------------------------------------------------
__global__ void pool_scatter_kernel(const int* __restrict__ batch,
                                    const float* __restrict__ h,
                                    float* __restrict__ g, int nNodes) {
  long long t = (long long)blockIdx.x * blockDim.x + threadIdx.x;
  long long total = (long long)nNodes * 32;
  if (t >= total) return;
  int v  = (int)(t >> 5);
  int c4 = (int)(t & 31) << 2;
  int gi = batch[v];
  float4 val = *reinterpret_cast<const float4*>(&h[(size_t)v * HID + c4]);
  float* p = &g[(size_t)gi * HID + c4];
  atomicAdd(p + 0, val.x);
  atomicAdd(p + 1, val.y);
  atomicAdd(p + 2, val.z);
  atomicAdd(p + 3, val.w);
}

// ---------------------------------------------------------------------------
// out = relu(in @ W + bias)   M x 128 @ 128 x 128, f32 via V_WMMA_F32_16X16X4_F32
// Block = 128 threads = 4 waves; each wave computes a 16-row x 128-col strip.
//
// W is staged TRANSPOSED in LDS (sWt[col][k], padded stride WSTRIDE) so the
// B-fragment {W[ka][col], W[ka+1][col]} is one contiguous 8B-aligned float2:
// a single ds_load_b64 into an even VGPR pair -> no v_dual_mov shuffling.
//
// A-frag layout (16x4 f32, ISA 7.12.2): lanes 0-15 rows M=0..15 with K={k0,k0+1},
// lanes 16-31 same rows with K={k0+2,k0+3} -> contiguous float2 per lane.
// C/D (16x16 f32): lane half selects M base (0/8), VGPR r -> M=r, col = lane&15.
// ---------------------------------------------------------------------------
__global__ __launch_bounds__(128) void gemm128_relu_wmma_kernel(
    const float* __restrict__ in, const float* __restrict__ W,
    const float* __restrict__ bias, float* __restrict__ out, int M) {
  __shared__ float sWt[HID * WSTRIDE];  // transposed weights, ~65KB
  __shared__ float sB[HID];

  const int tid = threadIdx.x;
  // Transpose-on-write staging: coalesced b128 global reads, b32 LDS writes.
  for (int i = tid * 4; i < HID * HID; i += 128 * 4) {
    float4 w = *reinterpret_cast<const float4*>(&W[i]);
    const int k = i >> 7;        // row of W (K index)
    const int c = i & (HID - 1); // first of 4 consecutive cols
    sWt[(c + 0) * WSTRIDE + k] = w.x;
    sWt[(c + 1) * WSTRIDE + k] = w.y;
    sWt[(c + 2) * WSTRIDE + k] = w.z;
    sWt[(c + 3) * WSTRIDE + k] = w.w;
  }
  sB[tid] = bias[tid];
  __syncthreads();

  const int wave = tid >> 5;
  const int lane = tid & 31;
  const int rowBase = blockIdx.x * 64 + wave * 16;
  if (rowBase >= M) return;  // wave-uniform: EXEC stays all-ones for WMMA
  const int lrow = lane & 15;
  const int lhi  = lane >> 4;  // 0 or 1

  v8f acc[8];
#pragma unroll
  for (int nt = 0; nt < 8; ++nt) acc[nt] = (v8f)(0.0f);

  const float* arow = in + (size_t)(rowBase + lrow) * HID;

  for (int k0 = 0; k0 < HID; k0 += 4) {
    const int ka = k0 + (lhi << 1);
    v2f a = *reinterpret_cast<const v2f*>(&arow[ka]);
#pragma unroll
    for (int nt = 0; nt < 8; ++nt) {
      const int col = nt * 16 + lrow;
      v2f b = *reinterpret_cast<const v2f*>(&sWt[col * WSTRIDE + ka]);
      acc[nt] = __builtin_amdgcn_wmma_f32_16x16x4_f32(
          /*neg_a=*/false, a, /*neg_b=*/false, b,
          /*c_mod=*/(short)0, acc[nt], /*reuse_a=*/false, /*reuse_b=*/false);
    }
  }

#pragma unroll
  for (int nt = 0; nt < 8; ++nt) {
    const int col = nt * 16 + lrow;
    const float bb = sB[col];
#pragma unroll
    for (int r = 0; r < 8; ++r) {
      const int row = rowBase + lhi * 8 + r;
      float val = acc[nt][r] + bb;
      out[(size_t)row * HID + col] = val > 0.0f ? val : 0.0f;
    }
  }
}

// ---------------------------------------------------------------------------
// Head: mu = g @ W_mu + b_mu ; logvar = g @ W_lv + b_lv  (1000x128 @ 128x64)
// One block per graph; g row staged in LDS. Tiny (16M FMA) -> scalar kernel.
// ---------------------------------------------------------------------------
__global__ __launch_bounds__(128) void head_kernel(
    const float* __restrict__ g, const float* __restrict__ Wmu,
    const float* __restrict__ bmu, const float* __restrict__ Wlv,
    const float* __restrict__ blv, float* __restrict__ out, int nGraphs) {
  __shared__ float sg[HID];
  int gi = blockIdx.x;
  int r  = threadIdx.x;  // 0..127
  sg[r] = g[(size_t)gi * HID + r];
  __syncthreads();

  const bool is_mu = (r < 64);
  const int j = is_mu ? r : (r - 64);
  const float* Wc = is_mu ? Wmu : Wlv;
  float s = is_mu ? bmu[j] : blv[j];
#pragma unroll 4
  for (int k = 0; k < HID; ++k) s += sg[k] * Wc[k * 64 + j];
  float* oc = is_mu ? out : (out + (size_t)nGraphs * 64);
  oc[(size_t)gi * 64 + j] = s;
}

// ---------------------------------------------------------------------------
extern "C" void kernel_launch(void* const* d_in, const int* in_sizes, int n_in,
                              void* d_out, int out_size, void* d_ws,
                              size_t ws_size, hipStream_t stream) {
  const float* x    = (const float*)d_in[0];
  const int*   ei   = (const int*)d_in[1];   // [2, NE] flat
  const int*   bat  = (const int*)d_in[2];
  const float* W1   = (const float*)d_in[3]; // [3,128,128]
  const float* b1   = (const float*)d_in[4]; // [3,128]
  const float* W2   = (const float*)d_in[5];
  const float* b2   = (const float*)d_in[6];
  const float* Wmu  = (const float*)d_in[7]; // [128,64]
  const float* bmu  = (const float*)d_in[8];
  const float* Wlv  = (const float*)d_in[9];
  const float* blv  = (const float*)d_in[10];
  float* out = (float*)d_out;

  const int M  = in_sizes[0] / HID;   // 100000 nodes
  const int NE = in_sizes[1] / 2;     // 1.6M edges
  const int NG = out_size / 128;      // 1000 graphs (2 x NG x 64 outputs)

  const int* src = ei;
  const int* dst = ei + NE;

  // Workspace layout: h | tmp | agg | g  (f32)
  const size_t nodeElems = (size_t)M * HID;
  float* hbuf = (float*)d_ws;
  float* tmp  = hbuf + nodeElems;
  float* agg  = tmp + nodeElems;
  float* gbuf = agg + nodeElems;

  const long long n4 = (long long)nodeElems / 4;
  const int B = 256;
  dim3 gridCopy((unsigned)((n4 + B - 1) / B));
  dim3 gridEdge((unsigned)(((long long)NE * 32 + B - 1) / B));
  dim3 gridGemm((unsigned)((M + 63) / 64));
  dim3 gridPool((unsigned)(((long long)M * 32 + B - 1) / B));

  const float* h_in = x;
  for (int layer = 0; layer < 3; ++layer) {
    // agg = h + scatter_add(h[src] -> dst)
    copy_f4_kernel<<<gridCopy, B, 0, stream>>>(h_in, agg, n4);
    edge_scatter_kernel<<<gridEdge, B, 0, stream>>>(src, dst, h_in, agg, NE);
    // h = relu(relu(agg @ W1 + b1) @ W2 + b2)
    gemm128_relu_wmma_kernel<<<gridGemm, 128, 0, stream>>>(
        agg, W1 + (size_t)layer * HID * HID, b1 + (size_t)layer * HID, tmp, M);
    gemm128_relu_wmma_kernel<<<gridGemm, 128, 0, stream>>>(
        tmp, W2 + (size_t)layer * HID * HID, b2 + (size_t)layer * HID, hbuf, M);
    h_in = hbuf;
  }

  // global_add_pool
  long long gElems = (long long)NG * HID;
  zero_kernel<<<(unsigned)((gElems + B - 1) / B), B, 0, stream>>>(gbuf, gElems);
  pool_scatter_kernel<<<gridPool, B, 0, stream>>>(bat, hbuf, gbuf, M);

  // mu / logvar heads
  head_kernel<<<NG, 128, 0, stream>>>(gbuf, Wmu, bmu, Wlv, blv, out, NG);
}